// LFQ_85040352461324
// MI455X (gfx1250) — compile-verified
//
#include <hip/hip_runtime.h>
#include <math.h>

typedef float v2f __attribute__((ext_vector_type(2)));
typedef float v8f __attribute__((ext_vector_type(8)));

#define S_TOTAL 8192
#define DIMV    512
#define CD      14
#define EPSV    1e-5f

// workspace layout (float offsets)
#define WS_Q     0                        // [8192][16] q = sigmoid(400h)
#define WS_A     131072                   // [128][8192] high-7-bit partial products (transposed: jh-major)
#define WS_B     (131072 + 1048576)       // [128][8192] low-7-bit partial products (jl-major)
#define WS_IDX   (131072 + 2*1048576)     // [8192] packed 14-bit indices (as uint)
#define WS_SCAL  (WS_IDX + 8192)          // [0]=per-sample-entropy, [1]=commit, [2]=codebook-entropy
#define WS_WPAD  (WS_SCAL + 4)            // [16][512] zero-padded W_in (16B-aligned offset)

// ---------------------------------------------------------------------------
// Init: zero scalar accumulators and build zero-padded W_in (rows 14,15 = 0)
// so the WMMA B-operand needs no per-iteration masking.
// ---------------------------------------------------------------------------
__global__ __launch_bounds__(512) void lfq_init_pad(const float* __restrict__ Win,
                                                    float* __restrict__ ws) {
  const int t = blockIdx.x * 512 + threadIdx.x;   // 0..8191
  const int row = t >> 9, col = t & 511;
  ws[WS_WPAD + t] = (row < CD) ? Win[row * DIMV + col] : 0.f;
  if (t < 3) ws[WS_SCAL + t] = 0.f;
}

// ---------------------------------------------------------------------------
// Phase 1: h = x*W_in^T + b_in via WMMA f32 16x16x4 (K-loop over 512).
// W_pad is staged into LDS with global_load_async_to_lds_b128 (ASYNCcnt path);
// B-fragments then come from LDS (ds_load_b64), A streams from global.
// Fused: ballot -> 14-bit indices, commit loss, per-sample entropy,
// q_k = sigmoid(400 h_k) stored for the avg_prob factorization.
// A layout: lane = (K>=2)*16 + M, vgpr = K%2. B mirrored with N = lane&15.
// C layout: vgpr r, lanes 0-15 -> M=r, lanes 16-31 -> M=8+r, N = lane&15.
// ---------------------------------------------------------------------------
__global__ __launch_bounds__(32) void lfq_project_in(
    const float* __restrict__ x, const float* __restrict__ bin,
    float* __restrict__ ws, float* __restrict__ idx_out)
{
  __shared__ float wlds[16 * DIMV];                // 32 KB padded W_in
  const int lane = threadIdx.x;
  const int half = lane >> 4;
  const int n    = lane & 15;
  const int s0   = blockIdx.x * 16;
  const bool nvalid = (n < CD);

  // ---- async copy W_pad (32 KB) from ws to LDS: 2048 x b128, 64 per lane ----
  const float* wpad = ws + WS_WPAD;
  for (int cc = 0; cc < 64; ++cc) {
    const int elt = (cc * 32 + lane) * 4;          // float index of 16B chunk
    unsigned lds_addr = (unsigned)(unsigned long long)(&wlds[elt]);
    unsigned long long gaddr = (unsigned long long)(const void*)(wpad + elt);
    asm volatile("global_load_async_to_lds_b128 %0, %1, off"
                 :: "v"(lds_addr), "v"(gaddr) : "memory");
  }
  asm volatile("s_wait_asynccnt 0x0" ::: "memory");

  const float* xrow = x + (size_t)(s0 + n) * DIMV + 2 * half;
  const float* wrow = &wlds[n * DIMV + 2 * half];  // rows 14,15 are zero

  v8f c = {0.f,0.f,0.f,0.f,0.f,0.f,0.f,0.f};
  for (int k = 0; k < DIMV; k += 4) {
    v2f a = *(const v2f*)(xrow + k);               // global_load_b64
    v2f b = *(const v2f*)(wrow + k);               // ds_load_b64
    c = __builtin_amdgcn_wmma_f32_16x16x4_f32(false, a, false, b, (short)0, c,
                                              false, false);
  }

  const float bias = nvalid ? bin[n] : 0.f;
  float ent_local = 0.f, commit_local = 0.f;
  unsigned* idx_store = (unsigned*)(ws + WS_IDX);
  float*    qarr      = ws + WS_Q;

  #pragma unroll
  for (int r = 0; r < 8; ++r) {
    float h = c[r] + bias;
    unsigned mask = (unsigned)__ballot(h > 0.f);   // bits[13:0]=row r, bits[29:16]=row r+8
    if (lane == 0) {
      unsigned idx_lo = __brev(mask & 0x3FFFu) >> 18;        // bit n -> weight 2^(13-n)
      unsigned idx_hi = __brev((mask >> 16) & 0x3FFFu) >> 18;
      idx_store[s0 + r]     = idx_lo;
      idx_store[s0 + 8 + r] = idx_hi;
      idx_out[s0 + r]       = (float)idx_lo;
      idx_out[s0 + 8 + r]   = (float)idx_hi;
    }
    if (nvalid) {
      const int s = s0 + half * 8 + r;
      float z  = 400.f * h;                 // 2 * INV_TEMPERATURE * 2h
      float az = fabsf(z);
      float t  = expf(-az);                 // [0,1], underflows cleanly to 0
      float q  = (z >= 0.f) ? (1.f / (1.f + t)) : (t / (1.f + t));
      qarr[(size_t)s * 16 + n] = q;
      ent_local += log1pf(t) + az * t / (1.f + t);  // stable binary entropy
      float dd = fabsf(h) - 1.f;                    // (h - sign(h))^2
      commit_local += dd * dd;
    }
  }
  if (nvalid) {
    atomicAdd(ws + WS_SCAL + 0, ent_local);
    atomicAdd(ws + WS_SCAL + 1, commit_local);
  }
}

// ---------------------------------------------------------------------------
// Phase 2a: per-sample 7-bit Kronecker partial products, stored transposed
// ([jh][s] / [jl][s]) so the GEMM's K-stream is contiguous per lane.
// j = jh*128 + jl ; p_j = A[jh][s] * B[jl][s]
// ---------------------------------------------------------------------------
__global__ __launch_bounds__(256) void lfq_build_ab(float* __restrict__ ws)
{
  __shared__ float qs[CD];
  const int s = blockIdx.x;
  const int t = threadIdx.x;
  if (t < CD) qs[t] = ws[WS_Q + (size_t)s * 16 + t];
  __syncthreads();
  if (t < 128) {
    float p = 1.f;
    #pragma unroll
    for (int nn = 0; nn < 7; ++nn) {        // bits 13..7 of j -> q_0..q_6
      float q = qs[nn];
      p *= ((t >> (6 - nn)) & 1) ? q : (1.f - q);
    }
    ws[WS_A + (size_t)t * S_TOTAL + s] = p;
  } else {
    const int jl = t - 128;
    float p = 1.f;
    #pragma unroll
    for (int nn = 7; nn < 14; ++nn) {       // bits 6..0 of j -> q_7..q_13
      float q = qs[nn];
      p *= ((jl >> (13 - nn)) & 1) ? q : (1.f - q);
    }
    ws[WS_B + (size_t)jl * S_TOTAL + s] = p;
  }
}

// ---------------------------------------------------------------------------
// Phase 2b: avg_prob = (1/N) * A B^T over K=8192 via WMMA f32 16x16x4,
// fused with the codebook-entropy reduction (avg_prob never materialized).
// A/B are [row][K] so each lane streams two contiguous b64 loads per step;
// the whole 8 MB A/B working set is L2-resident (192 MB L2).
// ---------------------------------------------------------------------------
__global__ __launch_bounds__(32) void lfq_codebook_entropy(float* __restrict__ ws)
{
  const int lane = threadIdx.x;
  const int half = lane >> 4;
  const int n    = lane & 15;
  const int jh   = (blockIdx.x >> 3) * 16 + n;   // M index for A loads
  const int jl   = (blockIdx.x & 7)  * 16 + n;   // N index for B loads
  const float* Arow = ws + WS_A + (size_t)jh * S_TOTAL + 2 * half;
  const float* Brow = ws + WS_B + (size_t)jl * S_TOTAL + 2 * half;

  v8f c = {0.f,0.f,0.f,0.f,0.f,0.f,0.f,0.f};
  for (int k = 0; k < S_TOTAL; k += 4) {
    v2f a = *(const v2f*)(Arow + k);
    v2f b = *(const v2f*)(Brow + k);
    c = __builtin_amdgcn_wmma_f32_16x16x4_f32(false, a, false, b, (short)0, c,
                                              false, false);
  }
  float e = 0.f;
  const float invN = 1.f / (float)S_TOTAL;
  #pragma unroll
  for (int r = 0; r < 8; ++r) {
    float ap = c[r] * invN;
    e += -ap * logf(fmaxf(ap, EPSV));       // matches -p*log(clip(p, EPS))
  }
  atomicAdd(ws + WS_SCAL + 2, e);
}

// ---------------------------------------------------------------------------
// Phase 3: out = sign(h)*W_out^T + b_out via WMMA (K=14 padded to 16).
// Signs reconstructed from the packed 14-bit index per sample.
// ---------------------------------------------------------------------------
__global__ __launch_bounds__(32) void lfq_project_out(
    const float* __restrict__ Wout, const float* __restrict__ bout,
    const float* __restrict__ ws, float* __restrict__ out)
{
  const int lane = threadIdx.x;
  const int half = lane >> 4;
  const int n    = lane & 15;
  const int s0   = blockIdx.y * 16;
  const int d0   = blockIdx.x * 16;
  const unsigned* idx_store = (const unsigned*)(ws + WS_IDX);
  const unsigned idxv = idx_store[s0 + n];   // sample row for A loads (M = lane&15)
  const int d = d0 + n;                      // output col for B loads

  v8f c = {0.f,0.f,0.f,0.f,0.f,0.f,0.f,0.f};
  #pragma unroll
  for (int kb = 0; kb < 16; kb += 4) {
    const int k0 = kb + 2 * half;
    const int k1 = k0 + 1;
    const int sh0 = (13 - k0) < 0 ? 0 : (13 - k0);
    const int sh1 = (13 - k1) < 0 ? 0 : (13 - k1);
    v2f a, b;
    a.x = (k0 < CD) ? (((idxv >> sh0) & 1u) ? 1.f : -1.f) : 0.f;
    a.y = (k1 < CD) ? (((idxv >> sh1) & 1u) ? 1.f : -1.f) : 0.f;
    b.x = (k0 < CD) ? Wout[(size_t)d * CD + k0] : 0.f;
    b.y = (k1 < CD) ? Wout[(size_t)d * CD + k1] : 0.f;
    c = __builtin_amdgcn_wmma_f32_16x16x4_f32(false, a, false, b, (short)0, c,
                                              false, false);
  }
  const float bb = bout[d];
  #pragma unroll
  for (int r = 0; r < 8; ++r) {
    const int s = s0 + half * 8 + r;
    out[(size_t)s * DIMV + d] = c[r] + bb;
  }
}

__global__ void lfq_finalize(const float* __restrict__ ws, float* __restrict__ aux_out)
{
  if (threadIdx.x == 0 && blockIdx.x == 0) {
    float per_sample = ws[WS_SCAL + 0] / (float)S_TOTAL;
    float commit     = ws[WS_SCAL + 1] / ((float)S_TOTAL * (float)CD);
    float cb_ent     = ws[WS_SCAL + 2];
    aux_out[0] = (per_sample - 1.0f * cb_ent) * 0.1f + commit * 0.25f;
  }
}

extern "C" void kernel_launch(void* const* d_in, const int* in_sizes, int n_in,
                              void* d_out, int out_size, void* d_ws, size_t ws_size,
                              hipStream_t stream) {
  const float* x    = (const float*)d_in[0];
  const float* Win  = (const float*)d_in[1];
  const float* bin  = (const float*)d_in[2];
  const float* Wout = (const float*)d_in[3];
  const float* bout = (const float*)d_in[4];
  // d_in[5] (codebook) unused: it is exactly the set of all 2^14 sign patterns.

  float* out     = (float*)d_out;
  float* idx_out = out + (size_t)S_TOTAL * DIMV;   // indices region (as float)
  float* aux_out = idx_out + S_TOTAL;              // scalar aux loss
  float* ws      = (float*)d_ws;

  lfq_init_pad<<<16, 512, 0, stream>>>(Win, ws);
  lfq_project_in<<<S_TOTAL / 16, 32, 0, stream>>>(x, bin, ws, idx_out);
  lfq_build_ab<<<S_TOTAL, 256, 0, stream>>>(ws);
  lfq_codebook_entropy<<<64, 32, 0, stream>>>(ws);
  lfq_project_out<<<dim3(DIMV / 16, S_TOTAL / 16), 32, 0, stream>>>(Wout, bout, ws, out);
  lfq_finalize<<<1, 32, 0, stream>>>(ws, aux_out);
}